// GravityODECell_52802327937325
// MI455X (gfx1250) — compile-verified
//
#include <hip/hip_runtime.h>

typedef float fv4 __attribute__((ext_vector_type(4)));

#define DTF 0.1f  // DT / N_STEPS = 0.5 / 5

// One explicit-Euler sub-step for a single 3-body/2D system held in registers.
// k = DTF * A (A folded once).
// Note: clamp(sqrt(clamp(d2,0.1,1e5)),1,170) == clamp(sqrt(d2),1,170) for d2>=0,
// so the inner clip is dropped (exact equivalence, sqrt is monotone).
__device__ __forceinline__ void body_step(float p[6], float v[6], float k) {
    float vx[3], vy[3];
    vx[0] = p[0] - p[2];  vy[0] = p[1] - p[3];   // body0 - body1
    vx[1] = p[2] - p[4];  vy[1] = p[3] - p[5];   // body1 - body2
    vx[2] = p[4] - p[0];  vy[2] = p[5] - p[1];   // body2 - body0

    float fx[3], fy[3];
#pragma unroll
    for (int i = 0; i < 3; ++i) {
        float d2 = fmaf(vx[i], vx[i], vy[i] * vy[i]);
        float n  = __builtin_amdgcn_sqrtf(d2);              // v_sqrt_f32 (1 ulp)
        float c  = fminf(fmaxf(n, 1.0f), 170.0f);           // clip(norm,1,170)
        float c3 = c * c * c;
        float inv = __builtin_amdgcn_rcpf(c3);              // v_rcp_f32 (1 ulp), c3 in [1, 4.9e6]
        fx[i] = vx[i] * inv;
        fy[i] = vy[i] * inv;
    }

    // F_i_new = -(A*(F_i - F_{i-1}))  ->  vel += k*(F_{i-1} - F_i),  k = DTF*A
    v[0] = fmaf(k, fx[2] - fx[0], v[0]);
    v[1] = fmaf(k, fy[2] - fy[0], v[1]);
    v[2] = fmaf(k, fx[0] - fx[1], v[2]);
    v[3] = fmaf(k, fy[0] - fy[1], v[3]);
    v[4] = fmaf(k, fx[1] - fx[2], v[4]);
    v[5] = fmaf(k, fy[1] - fy[2], v[5]);

#pragma unroll
    for (int j = 0; j < 6; ++j)
        p[j] = fmaf(DTF, v[j], p[j]);
}

// Vectorized path: 2 systems per thread = 3x b128 loads per array, 16B aligned.
__global__ __launch_bounds__(256) void grav_pairs_kernel(
    const fv4* __restrict__ poss, const fv4* __restrict__ vels,
    const float* __restrict__ Aptr,
    fv4* __restrict__ pout, fv4* __restrict__ vout,
    long long npairs)
{
    long long t = (long long)blockIdx.x * blockDim.x + threadIdx.x;
    if (t >= npairs) return;

    const float A = *Aptr;          // uniform -> s_load
    const float k = DTF * A;

    const long long b = 3 * t;
    fv4 pa = __builtin_nontemporal_load(poss + b + 0);
    fv4 pb = __builtin_nontemporal_load(poss + b + 1);
    fv4 pc = __builtin_nontemporal_load(poss + b + 2);
    fv4 va = __builtin_nontemporal_load(vels + b + 0);
    fv4 vb = __builtin_nontemporal_load(vels + b + 1);
    fv4 vc = __builtin_nontemporal_load(vels + b + 2);

    float p0[6] = {pa.x, pa.y, pa.z, pa.w, pb.x, pb.y};
    float p1[6] = {pb.z, pb.w, pc.x, pc.y, pc.z, pc.w};
    float v0[6] = {va.x, va.y, va.z, va.w, vb.x, vb.y};
    float v1[6] = {vb.z, vb.w, vc.x, vc.y, vc.z, vc.w};

#pragma unroll
    for (int s = 0; s < 5; ++s) {
        body_step(p0, v0, k);       // two independent chains -> ILP across
        body_step(p1, v1, k);       // v_sqrt/v_rcp latency
    }

    fv4 oa = {p0[0], p0[1], p0[2], p0[3]};
    fv4 ob = {p0[4], p0[5], p1[0], p1[1]};
    fv4 oc = {p1[2], p1[3], p1[4], p1[5]};
    __builtin_nontemporal_store(oa, pout + b + 0);
    __builtin_nontemporal_store(ob, pout + b + 1);
    __builtin_nontemporal_store(oc, pout + b + 2);

    fv4 wa = {v0[0], v0[1], v0[2], v0[3]};
    fv4 wb = {v0[4], v0[5], v1[0], v1[1]};
    fv4 wc = {v1[2], v1[3], v1[4], v1[5]};
    __builtin_nontemporal_store(wa, vout + b + 0);
    __builtin_nontemporal_store(wb, vout + b + 1);
    __builtin_nontemporal_store(wc, vout + b + 2);
}

// Scalar fallback (only used if B is odd; avoids misaligned b128).
__global__ __launch_bounds__(256) void grav_scalar_kernel(
    const float* __restrict__ poss, const float* __restrict__ vels,
    const float* __restrict__ Aptr,
    float* __restrict__ pout, float* __restrict__ vout,
    long long n)
{
    long long i = (long long)blockIdx.x * blockDim.x + threadIdx.x;
    if (i >= n) return;

    const float A = *Aptr;
    const float k = DTF * A;

    float p[6], v[6];
#pragma unroll
    for (int j = 0; j < 6; ++j) {
        p[j] = poss[6 * i + j];
        v[j] = vels[6 * i + j];
    }
#pragma unroll
    for (int s = 0; s < 5; ++s)
        body_step(p, v, k);
#pragma unroll
    for (int j = 0; j < 6; ++j) {
        pout[6 * i + j] = p[j];
        vout[6 * i + j] = v[j];
    }
}

extern "C" void kernel_launch(void* const* d_in, const int* in_sizes, int n_in,
                              void* d_out, int out_size, void* d_ws, size_t ws_size,
                              hipStream_t stream) {
    (void)n_in; (void)out_size; (void)d_ws; (void)ws_size;

    const long long B = (long long)in_sizes[0] / 6;   // 4,194,304
    const float* Aptr = (const float*)d_in[2];

    float* out  = (float*)d_out;
    float* pout = out;            // first B*6 floats: positions
    float* vout = out + B * 6;    // next  B*6 floats: velocities

    const int block = 256;

    if ((B & 1LL) == 0) {
        const long long npairs = B >> 1;               // 2,097,152 threads
        const long long grid = (npairs + block - 1) / block;   // 8192 blocks
        grav_pairs_kernel<<<(unsigned int)grid, block, 0, stream>>>(
            (const fv4*)d_in[0], (const fv4*)d_in[1], Aptr,
            (fv4*)pout, (fv4*)vout, npairs);
    } else {
        const long long grid = (B + block - 1) / block;
        grav_scalar_kernel<<<(unsigned int)grid, block, 0, stream>>>(
            (const float*)d_in[0], (const float*)d_in[1], Aptr,
            pout, vout, B);
    }
}